// radon_55662776156417
// MI455X (gfx1250) — compile-verified
//
#include <hip/hip_runtime.h>

typedef __attribute__((ext_vector_type(2))) float v2f;
typedef __attribute__((ext_vector_type(8))) float v8f;

#define N_ANGLES 512
#define N_DET    512
#define IMG_W    512
#define IMG_H    512
#define N_SAMP   256

__global__ __launch_bounds__(256) void radon_fwd_kernel(const float* __restrict__ img,
                                                        float* __restrict__ out)
{
    const int lane = threadIdx.x;                 // 0..31
    const int wv   = threadIdx.y;                 // 0..7 (wave in block)
    const int a    = blockIdx.y;                  // angle index (pre-flip)
    const int d0   = (blockIdx.x * 8 + wv) * 16;  // first detector of this wave
    const int n    = lane & 15;                   // detector column N = lane % 16
    const int d    = d0 + n;
    const int hi   = lane >> 4;                   // which half of the wave

    // ---- per-ray setup (mirrors reference f32 math) ----
    const float PI = 3.14159265358979323846f;     // rounds to f32 pi like np.pi
    float theta = (float)a * (PI / 511.0f) + 0.5f * PI;
    float ca, sa;
    sincosf(theta, &sa, &ca);

    const float sy = -1.41421356237309504880f;    // -sqrt(2) in f32
    // sx = linspace(-1,1,512)[d] ; the (det/2)*distance*step factor == 1.0 exactly
    float sx = fmaf((float)d, 2.0f / 511.0f, -1.0f);

    float rsx = sx * ca - sy * sa;
    float rsy = sx * sa + sy * ca;
    float rex = sx * ca + sy * sa;
    float rey = sx * sa - sy * ca;
    float rdx = rex - rsx;
    float rdy = rey - rsy;

    // bilinear sample at parameter s/256 along the ray (zeros padding, align_corners)
    auto sample = [&](int s) -> float {
        float t  = (float)s * (1.0f / (float)N_SAMP);   // exact: s/256
        float gx = rsx + t * rdx;
        float gy = rsy + t * rdy;
        float x  = ((gx + 1.0f) * 0.5f) * (float)(IMG_W - 1);
        float y  = ((gy + 1.0f) * 0.5f) * (float)(IMG_H - 1);
        float x0 = floorf(x);
        float y0 = floorf(y);
        float wx1 = x - x0, wx0 = 1.0f - wx1;
        float wy1 = y - y0, wy0 = 1.0f - wy1;
        float x1 = x0 + 1.0f, y1 = y0 + 1.0f;

        float mx0 = (x0 >= 0.0f && x0 <= 511.0f) ? 1.0f : 0.0f;
        float mx1 = (x1 >= 0.0f && x1 <= 511.0f) ? 1.0f : 0.0f;
        float my0 = (y0 >= 0.0f && y0 <= 511.0f) ? 1.0f : 0.0f;
        float my1 = (y1 >= 0.0f && y1 <= 511.0f) ? 1.0f : 0.0f;

        int ix0 = (int)fminf(fmaxf(x0, 0.0f), 511.0f);
        int ix1 = (int)fminf(fmaxf(x1, 0.0f), 511.0f);
        int iy0 = (int)fminf(fmaxf(y0, 0.0f), 511.0f);
        int iy1 = (int)fminf(fmaxf(y1, 0.0f), 511.0f);

        float v00 = img[iy0 * IMG_W + ix0];
        float v01 = img[iy0 * IMG_W + ix1];
        float v10 = img[iy1 * IMG_W + ix0];
        float v11 = img[iy1 * IMG_W + ix1];

        float r0 = (v00 * (wx0 * mx0)) + (v01 * (wx1 * mx1));
        float r1 = (v10 * (wx0 * mx0)) + (v11 * (wx1 * mx1));
        return r0 * (wy0 * my0) + r1 * (wy1 * my1);
    };

    // ---- WMMA reduction over sampling points ----
    // A = ones(16x4)  =>  D[M,N] = sum_k B[k,N] + C[M,N]  (every row identical).
    // Each lane supplies 2 of the 4 k-slots for its detector column N = lane%16;
    // which half supplies which k is irrelevant because A is all-ones.
    v2f ones = {1.0f, 1.0f};
    v8f acc  = {};                                // C starts at zero
    const int sbase = hi * 2;

#pragma unroll 2
    for (int it = 0; it < N_SAMP / 4; ++it) {
        int s = it * 4 + sbase;
        v2f b;
        b[0] = sample(s);
        b[1] = sample(s + 1);
        acc = __builtin_amdgcn_wmma_f32_16x16x4_f32(
            /*neg_a=*/false, ones, /*neg_b=*/false, b,
            /*c_mod=*/(short)0, acc, /*reuse_a=*/false, /*reuse_b=*/false);
    }

    // Every row of D holds the full sum; lane L's acc[0] is element (M, N=L%16).
    // Lanes 0..15 store (upper half holds duplicates). Fold in the /S and the
    // double flip: out[511-a, 511-d] = intens[a, d].
    if (lane < 16) {
        int ao = (N_ANGLES - 1) - a;
        int dO = (N_DET - 1) - d;
        out[ao * N_DET + dO] = acc[0] * (1.0f / (float)N_SAMP);
    }
}

extern "C" void kernel_launch(void* const* d_in, const int* in_sizes, int n_in,
                              void* d_out, int out_size, void* d_ws, size_t ws_size,
                              hipStream_t stream) {
    (void)in_sizes; (void)n_in; (void)d_ws; (void)ws_size; (void)out_size;
    const float* img = (const float*)d_in[0];
    float* out = (float*)d_out;

    dim3 block(32, 8);                 // 8 wave32s -> 128 detectors of one angle
    dim3 grid(N_DET / (16 * 8), N_ANGLES);  // (4, 512)
    radon_fwd_kernel<<<grid, block, 0, stream>>>(img, out);
}